// DCWTv2InferenceCache_5111011082701
// MI455X (gfx1250) — compile-verified
//
#include <hip/hip_runtime.h>
#include <math.h>

#define NUM_HEADS   16
#define HEAD_DIM    64
#define K_MAX       64
#define LOCAL_WIN   512
#define LOG_N       17
#define LEAF_START  (1u << LOG_N)      // 131072
#define MAX_LEN     65536
#define MAX_NODES   18
#define MAXP        40                 // max partial slots (<= 1024/64 + 18)
#define TOK_STRIDE  1024               // NUM_HEADS*HEAD_DIM
#define FNODE_F     65536              // 64 * 1024 floats per node f-buffer
#define FNODE_V4    16384              // float4s per node f-buffer

typedef __attribute__((ext_vector_type(2))) float v2f;
typedef __attribute__((ext_vector_type(4))) float v4f;
typedef __attribute__((ext_vector_type(8))) float v8f;

// ---- workspace layout ----
// ints (from (int*)ws):
//   [0] = node count, [1] = total partial slots
//   node records at [2 + 6*i]: {a, L, depth, K, nch, pstart}
//   slot table at   [110 + 2*p]: {node, c0}
#define NODE_REC_OFF 2
#define SLOT_OFF     (NODE_REC_OFF + 6*MAX_NODES)     // 110 .. 189
// floats (from (float*)ws):
#define WS_SCALE_OFF   256                               // 32 floats
#define WS_QDEP_OFF    288                               // 18*1024
#define WS_NODEOUT_OFF (WS_QDEP_OFF + MAX_NODES*1024)    // 18*1024
#define WS_FBUF_OFF    (WS_NODEOUT_OFF + MAX_NODES*1024) // 18*65536
#define WS_PARTIAL_OFF (WS_FBUF_OFF + MAX_NODES*FNODE_F) // 40*65536

// -------------------------------------------------------------------------
// Kernel 1: device-side cover set + partial-slot table, per-depth scales,
// q_depth via f32 WMMA. Block = 256 threads = 8 waves; wave w -> depths
// {w, w+8, w+16} (wave-uniform loop keeps EXEC all-ones for WMMA).
// -------------------------------------------------------------------------
__global__ void dcwt_setup_kernel(const float* __restrict__ q,
                                  const float* __restrict__ w,
                                  const float* __restrict__ temp,
                                  const int*   __restrict__ n_tokens,
                                  int* ws_i, float* ws_f) {
  const int tid = threadIdx.x;

  if (tid == 0) {
    int pos = n_tokens[0];
    int cnt = 0, ptot = 0;
    if (pos > 0) {
      int p = pos < MAX_LEN ? pos : MAX_LEN;
      unsigned l = LEAF_START, r = LEAF_START + (unsigned)p;
      while (l < r) {
        unsigned nodes2[2]; int nn = 0;
        if (l & 1u) { nodes2[nn++] = l; l++; }
        if (r & 1u) { r--; nodes2[nn++] = r; }
        for (int j = 0; j < nn; ++j) {
          unsigned nd = nodes2[j];
          int dep = LOG_N - (31 - __clz((int)nd));
          int L   = 1 << dep;
          int nch = (L > K_MAX) ? (L >> 6) : 1;
          int pcount = (nch + 63) >> 6;
          int* rec = ws_i + NODE_REC_OFF + 6*cnt;
          rec[0] = (int)((nd << dep) - LEAF_START);    // a
          rec[1] = L;
          rec[2] = dep;
          rec[3] = L < K_MAX ? L : K_MAX;              // K
          rec[4] = nch;
          rec[5] = ptot;                               // pstart
          for (int s = 0; s < pcount; ++s) {
            ws_i[SLOT_OFF + 2*(ptot + s) + 0] = cnt;
            ws_i[SLOT_OFF + 2*(ptot + s) + 1] = s * 64;  // c0
          }
          ptot += pcount;
          cnt++;
        }
        l >>= 1; r >>= 1;
      }
    }
    ws_i[0] = cnt;
    ws_i[1] = ptot;
  }

  if (tid <= LOG_N) {                       // 18 depth scales
    float t  = temp[tid];
    float sp = log1pf(expf(t));             // softplus (temps are small)
    ws_f[WS_SCALE_OFF + tid] = 1.0f / ((sp + 1e-6f) * 8.0f);  // 8 = sqrt(64)
  }

  // q_depth[d] = q + q @ W_d^T   (16x64 = 16x64 * 64x64), exact f32 WMMA.
  const int lane = tid & 31;
  const int wave = tid >> 5;
  const int m16  = lane & 15;               // M (A) / N (B,D) sub-index
  const int khi  = (lane >> 4) * 2;         // K offset within 4-wide K step
  float* qdep = ws_f + WS_QDEP_OFF;

  // Preload all 16 A fragments (Q rows) once.
  v2f areg[16];
#pragma unroll
  for (int kk = 0; kk < 16; ++kk) {
    const int k = kk * 4 + khi;
    areg[kk].x = q[m16 * HEAD_DIM + k];
    areg[kk].y = q[m16 * HEAD_DIM + k + 1];
  }

  for (int d = wave; d <= LOG_N; d += 8) {
    const float* W = w + d * (HEAD_DIM * HEAD_DIM);
    for (int n0 = 0; n0 < HEAD_DIM; n0 += 16) {
      const int n = n0 + m16;
      // Batch-load all 16 B fragments (W^T tile) so loads pipeline.
      v2f breg[16];
#pragma unroll
      for (int kk = 0; kk < 16; ++kk) {
        const int k = kk * 4 + khi;
        breg[kk].x = W[n * HEAD_DIM + k];
        breg[kk].y = W[n * HEAD_DIM + k + 1];
      }
      v8f acc = {};
#pragma unroll
      for (int kk = 0; kk < 16; ++kk)
        acc = __builtin_amdgcn_wmma_f32_16x16x4_f32(
            false, areg[kk], false, breg[kk], (short)0, acc, false, false);
      // D layout: VGPR j -> M = j + 8*(lane>>4), N = n0 + (lane&15)
      const int mbase = (lane >> 4) * 8;
#pragma unroll
      for (int j = 0; j < 8; ++j) {
        const int m = mbase + j;
        qdep[d * 1024 + m * HEAD_DIM + n] = q[m * HEAD_DIM + n] + acc[j];
      }
    }
  }
}

// -------------------------------------------------------------------------
// Kernel 2a: stage-A partial sums. One partial slot = up to 64 chunks of a
// node. Grid: 64 blocks/slot x 256 threads; thread owns one float4 output.
// Chunk loop unrolled x8 with batched independent nontemporal b128 loads
// to keep ~8 loads in flight per wave (latency hiding for 23.3 TB/s).
// -------------------------------------------------------------------------
__global__ void dcwt_partial_sum_kernel(const float* __restrict__ v,
                                        const int* __restrict__ ws_i,
                                        float* ws_f) {
  const int p   = blockIdx.x >> 6;
  const int sub = blockIdx.x & 63;
  const int ptot = ws_i[1];
  if (p >= ptot) return;

  const int node = ws_i[SLOT_OFF + 2*p + 0];
  const int c0   = ws_i[SLOT_OFF + 2*p + 1];
  const int* rec = ws_i + NODE_REC_OFF + 6*node;
  const int a   = rec[0];
  const int K   = rec[3];
  const int nch = rec[4];
  const int cend = (c0 + 64 < nch) ? (c0 + 64) : nch;

  const int e4 = sub * 256 + threadIdx.x;   // float4 index within node tile
  const int k  = e4 >> 8;                   // 256 float4 per k-slot
  if (k >= K) return;

  const v4f* vp = (const v4f*)v;
  const long base = (long)a * (TOK_STRIDE / 4) + e4;  // +16384 per chunk

  v4f acc0 = {}, acc1 = {};
  int c = c0;
  for (; c + 8 <= cend; c += 8) {
    const v4f* pc = vp + base + (long)c * FNODE_V4;
    v4f x0 = __builtin_nontemporal_load(pc + 0L * FNODE_V4);
    v4f x1 = __builtin_nontemporal_load(pc + 1L * FNODE_V4);
    v4f x2 = __builtin_nontemporal_load(pc + 2L * FNODE_V4);
    v4f x3 = __builtin_nontemporal_load(pc + 3L * FNODE_V4);
    v4f x4 = __builtin_nontemporal_load(pc + 4L * FNODE_V4);
    v4f x5 = __builtin_nontemporal_load(pc + 5L * FNODE_V4);
    v4f x6 = __builtin_nontemporal_load(pc + 6L * FNODE_V4);
    v4f x7 = __builtin_nontemporal_load(pc + 7L * FNODE_V4);
    acc0 += (x0 + x2) + (x4 + x6);
    acc1 += (x1 + x3) + (x5 + x7);
  }
  for (; c < cend; ++c)
    acc0 += __builtin_nontemporal_load(vp + base + (long)c * FNODE_V4);

  ((v4f*)(ws_f + WS_PARTIAL_OFF))[(long)p * FNODE_V4 + e4] = acc0 + acc1;
}

// -------------------------------------------------------------------------
// Kernel 2b: stage-B combine. Sum a node's <=16 partial slots, scale by
// 1/nch, write f[node][k][h][d]. ~9 MB traffic, L2-resident.
// -------------------------------------------------------------------------
__global__ void dcwt_mean_combine_kernel(const int* __restrict__ ws_i,
                                         float* ws_f) {
  const int node = blockIdx.x >> 6;
  const int sub  = blockIdx.x & 63;
  const int cnt  = ws_i[0];
  if (node >= cnt) return;

  const int* rec = ws_i + NODE_REC_OFF + 6*node;
  const int K      = rec[3];
  const int nch    = rec[4];
  const int pstart = rec[5];
  const int pcount = (nch + 63) >> 6;

  const int e4 = sub * 256 + threadIdx.x;
  const int k  = e4 >> 8;
  if (k >= K) return;

  const v4f* part = (const v4f*)(ws_f + WS_PARTIAL_OFF);
  v4f acc = {};
  for (int s = 0; s < pcount; ++s)
    acc += part[(long)(pstart + s) * FNODE_V4 + e4];
  acc *= (1.0f / (float)nch);
  ((v4f*)(ws_f + WS_FBUF_OFF))[(long)node * FNODE_V4 + e4] = acc;
}

// -------------------------------------------------------------------------
// Kernel 3: per-(node, head) attention over <=64 keys. 64 threads (2 waves).
// -------------------------------------------------------------------------
__global__ void dcwt_node_attn_kernel(const int* __restrict__ ws_i,
                                      float* ws_f) {
  const int node = blockIdx.x >> 4;
  const int h    = blockIdx.x & 15;
  const int cnt  = ws_i[0];
  if (node >= cnt) return;

  const int* rec = ws_i + NODE_REC_OFF + 6*node;
  const int dep = rec[2];
  const int K   = rec[3];
  const float scale = ws_f[WS_SCALE_OFF + dep];
  const float* f  = ws_f + WS_FBUF_OFF + (long)node * FNODE_F + h * HEAD_DIM;
  const float* qd = ws_f + WS_QDEP_OFF + dep * 1024 + h * HEAD_DIM;

  __shared__ float sh_q[64];
  __shared__ float sh_w[64];
  __shared__ float red[64];
  const int t = threadIdx.x;

  sh_q[t] = qd[t];
  __syncthreads();

  float s = -INFINITY;
  if (t < K) {
    float acc = 0.f;
#pragma unroll 8
    for (int d2 = 0; d2 < HEAD_DIM; ++d2) acc += sh_q[d2] * f[t * TOK_STRIDE + d2];
    s = acc * scale;
  }

  red[t] = s; __syncthreads();
  for (int st = 32; st > 0; st >>= 1) { if (t < st) red[t] = fmaxf(red[t], red[t + st]); __syncthreads(); }
  const float mx = red[0]; __syncthreads();

  const float e = (t < K) ? expf(s - mx) : 0.f;
  red[t] = e; __syncthreads();
  for (int st = 32; st > 0; st >>= 1) { if (t < st) red[t] += red[t + st]; __syncthreads(); }
  const float tot = red[0]; __syncthreads();

  sh_w[t] = e / tot;
  __syncthreads();

  float out = 0.f;
  for (int k = 0; k < K; ++k) out += sh_w[k] * f[k * TOK_STRIDE + t];
  ws_f[WS_NODEOUT_OFF + node * 1024 + h * HEAD_DIM + t] = out;
}

// -------------------------------------------------------------------------
// Kernel 4: local-window attention (<=512 keys) + tree mean + final output.
// One block per head, 512 threads.
// -------------------------------------------------------------------------
__global__ void dcwt_local_combine_kernel(const float* __restrict__ v,
                                          const float* __restrict__ q,
                                          const int* __restrict__ n_tokens,
                                          const int* __restrict__ ws_i,
                                          const float* __restrict__ ws_f,
                                          float* __restrict__ out) {
  const int h = blockIdx.x;
  const int t = threadIdx.x;             // 0..511
  const int pos  = n_tokens[0];
  const int nloc = pos < LOCAL_WIN ? pos : LOCAL_WIN;
  const int start = pos - nloc;

  __shared__ float sh_q[64];
  __shared__ float sh_w[512];
  __shared__ float red[512];

  if (t < HEAD_DIM) sh_q[t] = q[h * HEAD_DIM + t];
  __syncthreads();

  float s = -INFINITY;
  if (t < nloc) {
    const float* vp = v + (long)(start + t) * TOK_STRIDE + h * HEAD_DIM;
    float acc = 0.f;
#pragma unroll 8
    for (int d = 0; d < HEAD_DIM; ++d) acc += sh_q[d] * vp[d];
    s = acc * 0.125f;                    // 1/sqrt(64)
  }

  red[t] = s; __syncthreads();
  for (int st = 256; st > 0; st >>= 1) { if (t < st) red[t] = fmaxf(red[t], red[t + st]); __syncthreads(); }
  const float mx = red[0]; __syncthreads();

  const float e = (t < nloc) ? expf(s - mx) : 0.f;
  red[t] = e; __syncthreads();
  for (int st = 256; st > 0; st >>= 1) { if (t < st) red[t] += red[t + st]; __syncthreads(); }
  const float tot = red[0]; __syncthreads();

  sh_w[t] = (tot > 0.f) ? e / tot : 0.f;
  __syncthreads();

  if (t < HEAD_DIM) {
    float acc = 0.f;
    for (int k = 0; k < nloc; ++k)
      acc += sh_w[k] * v[(long)(start + k) * TOK_STRIDE + h * HEAD_DIM + t];

    const int cnt = ws_i[0];
    float tsum = 0.f;
    for (int nd = 0; nd < cnt; ++nd)
      tsum += ws_f[WS_NODEOUT_OFF + nd * 1024 + h * HEAD_DIM + t];
    const float tree = cnt ? tsum / (float)cnt : 0.f;

    out[h * HEAD_DIM + t] = acc + tree;
  }
}

// -------------------------------------------------------------------------
extern "C" void kernel_launch(void* const* d_in, const int* in_sizes, int n_in,
                              void* d_out, int out_size, void* d_ws, size_t ws_size,
                              hipStream_t stream) {
  const float* v    = (const float*)d_in[0];   // (50000, 16, 64) f32
  const float* q    = (const float*)d_in[1];   // (1, 16, 64) f32
  const float* w    = (const float*)d_in[2];   // (18, 64, 64) f32
  const float* temp = (const float*)d_in[3];   // (18,) f32
  const int*   ntok = (const int*)d_in[4];     // scalar

  int*   ws_i = (int*)d_ws;
  float* ws_f = (float*)d_ws;

  dcwt_setup_kernel<<<1, 256, 0, stream>>>(q, w, temp, ntok, ws_i, ws_f);
  dcwt_partial_sum_kernel<<<MAXP * 64, 256, 0, stream>>>(v, ws_i, ws_f);
  dcwt_mean_combine_kernel<<<MAX_NODES * 64, 256, 0, stream>>>(ws_i, ws_f);
  dcwt_node_attn_kernel<<<MAX_NODES * 16, 64, 0, stream>>>(ws_i, ws_f);
  dcwt_local_combine_kernel<<<NUM_HEADS, 512, 0, stream>>>(v, q, ntok, ws_i, ws_f,
                                                           (float*)d_out);
}